// CausE_70351564308610
// MI455X (gfx1250) — compile-verified
//
#include <hip/hip_runtime.h>
#include <math.h>

// ---------------------------------------------------------------------------
// CausE scoring kernel for MI455X (gfx1250, wave32).
//
// preds[r] = f( dot(user_e[user[r]], item_e_c[item[r]]) ) for 1M rows, D=64.
// Memory-bound gather workload (~24us HBM floor at 23.3 TB/s). The 64-dim
// fp32 dot products for each group of 16 rows are computed as diag(U * V^T)
// using V_WMMA_F32_16X16X4_F32 (dense fp32 WMMA), accumulating 16 K=4 steps.
// ---------------------------------------------------------------------------

typedef float v2f __attribute__((ext_vector_type(2)));
typedef float v4f __attribute__((ext_vector_type(4)));
typedef float v8f __attribute__((ext_vector_type(8)));

#define EMBED 64

__device__ __forceinline__ float cause_epilogue(float pred, float popv,
                                                float ub, float ib) {
    // piecewise: exp(p) if p <= 0 else p + 1   -> result is always > 0
    pred = (pred <= 0.0f) ? expf(pred) : (pred + 1.0f);
    // popularity weight: pop^0.5
    float x = pred * sqrtf(popv);
    // log_sigmoid(x) with x > 0: -log1p(exp(-x)) (no overflow possible)
    float ls = -log1pf(expf(-x));
    return ls + ub + ib;
}

__global__ __launch_bounds__(256) void cause_wmma_kernel(
    const int* __restrict__ user, const int* __restrict__ item,
    const float* __restrict__ user_e, const float* __restrict__ item_e,
    const float* __restrict__ user_b, const float* __restrict__ item_b,
    const float* __restrict__ pop, float* __restrict__ out, int n_groups)
{
    const int lane = threadIdx.x & 31;
    const int wave = threadIdx.x >> 5;
    const int wavesPerBlock = blockDim.x >> 5;
    const int group = blockIdx.x * wavesPerBlock + wave;
    if (group >= n_groups) return;          // wave-uniform: EXEC stays all-1s

    const int m    = lane & 15;             // row within the 16-row group
    const int half = lane >> 4;             // 0: K pair {0,1}, 1: K pair {2,3}
    const int row  = group * 16 + m;

    const int uidx = user[row];             // lanes 16-31 re-load same 16 ints
    const int iidx = item[row];

    const float* __restrict__ uptr = user_e + (size_t)uidx * EMBED;
    const float* __restrict__ vptr = item_e + (size_t)iidx * EMBED;

    v8f acc = {0.f, 0.f, 0.f, 0.f, 0.f, 0.f, 0.f, 0.f};

    // A (16x4 fp32) and B (4x16 fp32) fragments share the same lane layout:
    // lane l holds two consecutive K values of row (l%16), lanes 16-31 hold
    // the next K pair. Dot products are invariant under K permutations applied
    // identically to A and B, so each lane pulls one float4 (b128) per step:
    //   lanes 0-15 : cols 8c .. 8c+3     lanes 16-31: cols 8c+4 .. 8c+7
    // and feeds (x,y) to WMMA #1 and (z,w) to WMMA #2. Every element of the
    // 16x64 U and V tiles is loaded exactly once (512B per load instruction).
    #pragma unroll
    for (int c = 0; c < 8; ++c) {
        const int col = c * 8 + half * 4;
        v4f a4 = *(const v4f* __restrict__)(uptr + col);
        v4f b4 = *(const v4f* __restrict__)(vptr + col);
        v2f a0 = { a4.x, a4.y }, a1 = { a4.z, a4.w };
        v2f b0 = { b4.x, b4.y }, b1 = { b4.z, b4.w };
        acc = __builtin_amdgcn_wmma_f32_16x16x4_f32(
                  false, a0, false, b0, (short)0, acc, false, false);
        acc = __builtin_amdgcn_wmma_f32_16x16x4_f32(
                  false, a1, false, b1, (short)0, acc, false, false);
    }

    // Extract diagonal: D[m][m] lives at VGPR (m%8), lane (m < 8 ? m : m+16).
    float pred = 0.0f;
    #pragma unroll
    for (int mm = 0; mm < 16; ++mm) {
        const int srcLane = (mm < 8) ? mm : (mm + 16);
        float dm = __shfl(acc[mm & 7], srcLane, 32);
        if (m == mm) pred = dm;             // lanes mm and mm+16 both take it
    }

    const float r = cause_epilogue(pred, pop[iidx], user_b[uidx], item_b[iidx]);
    if (half == 0) out[row] = r;            // one writer per row
}

// Scalar fallback for a batch tail (< 16 rows); unused when BATCH % 16 == 0.
__global__ void cause_tail_kernel(
    const int* __restrict__ user, const int* __restrict__ item,
    const float* __restrict__ user_e, const float* __restrict__ item_e,
    const float* __restrict__ user_b, const float* __restrict__ item_b,
    const float* __restrict__ pop, float* __restrict__ out, int start, int n)
{
    const int r = start + blockIdx.x * blockDim.x + threadIdx.x;
    if (r >= n) return;
    const int uidx = user[r];
    const int iidx = item[r];
    const float* __restrict__ up = user_e + (size_t)uidx * EMBED;
    const float* __restrict__ vp = item_e + (size_t)iidx * EMBED;
    float s = 0.0f;
    #pragma unroll
    for (int k = 0; k < EMBED; ++k) s = fmaf(up[k], vp[k], s);
    out[r] = cause_epilogue(s, pop[iidx], user_b[uidx], item_b[iidx]);
}

extern "C" void kernel_launch(void* const* d_in, const int* in_sizes, int n_in,
                              void* d_out, int out_size, void* d_ws, size_t ws_size,
                              hipStream_t stream) {
    const int*   user   = (const int*)d_in[0];
    const int*   item   = (const int*)d_in[1];
    const float* user_e = (const float*)d_in[2];
    const float* item_e = (const float*)d_in[3];
    const float* user_b = (const float*)d_in[4];
    const float* item_b = (const float*)d_in[5];
    const float* pop    = (const float*)d_in[6];
    float*       out    = (float*)d_out;

    const int n        = in_sizes[0];       // BATCH (= 1,048,576)
    const int n_groups = n / 16;
    const int rem      = n - n_groups * 16;

    if (n_groups > 0) {
        const int wavesPerBlock = 8;        // 256 threads = 8 wave32
        const int blocks = (n_groups + wavesPerBlock - 1) / wavesPerBlock;
        cause_wmma_kernel<<<blocks, 256, 0, stream>>>(
            user, item, user_e, item_e, user_b, item_b, pop, out, n_groups);
    }
    if (rem > 0) {
        cause_tail_kernel<<<1, 256, 0, stream>>>(
            user, item, user_e, item_e, user_b, item_b, pop, out,
            n_groups * 16, n);
    }
}